// GCN_58772332479092
// MI455X (gfx1250) — compile-verified
//
#include <hip/hip_runtime.h>
#include <hip/hip_bf16.h>

typedef float v2f __attribute__((ext_vector_type(2)));
typedef float v4f __attribute__((ext_vector_type(4)));
typedef float v8f __attribute__((ext_vector_type(8)));

#define NEG_SLOPE 0.1f

// ---------------------------------------------------------------------------
// Degree / normalization kernels
// ---------------------------------------------------------------------------
__global__ void deg_init_kernel(float* __restrict__ deg, int N) {
    int i = blockIdx.x * blockDim.x + threadIdx.x;
    if (i < N) deg[i] = 1.0f;   // self-loop contributes 1 to every node's degree
}

__global__ void deg_count_kernel(const long long* __restrict__ dst,
                                 float* __restrict__ deg, long long E) {
    long long i = (long long)blockIdx.x * blockDim.x + threadIdx.x;
    if (i < E) {
        long long d = dst[i];
        atomicAdd(&deg[d], 1.0f);
    }
}

__global__ void deg_rsqrt_kernel(float* __restrict__ deg, int N) {
    int i = blockIdx.x * blockDim.x + threadIdx.x;
    if (i < N) deg[i] = rsqrtf(deg[i]);   // deg >= 1 guaranteed (self-loops)
}

// ---------------------------------------------------------------------------
// WMMA f32 GEMM:  H[N x 64] = X[N x K] @ W[K x 64]
// 256 threads = 8 waves; each wave owns a 16-row strip and all 64 columns
// (4 accumulator tiles of 16x16).  K in {64, 128}.
//
// W is staged in LDS in k-pair-interleaved layout:
//   sW[(k>>1)*128 + n*2 + (k&1)]
// so a B fragment {W[kk][n], W[kk+1][n]} is ONE aligned ds_load_b64 into an
// even VGPR pair (no v_mov repacking).  X staged per-wave in 16x32 chunks.
// ---------------------------------------------------------------------------
__global__ __launch_bounds__(256)
void gcn_gemm_wmma_kernel(const float* __restrict__ X,
                          const float* __restrict__ Wm,
                          float* __restrict__ H, int N, int K) {
    __shared__ float sW[128 * 64];           // up to 32 KB, interleaved layout
    __shared__ float sX[8][16 * 32];         // 2 KB per wave

    const int tid   = threadIdx.x;
    const int wave  = tid >> 5;
    const int lane  = tid & 31;
    const int lrow  = lane & 15;             // M index (A/D) or N index (B/D)
    const int lhalf = lane >> 4;             // selects K pair / upper M half

    // stage W (block-wide, coalesced reads, interleaved LDS writes)
    for (int i = tid; i < K * 64; i += 256) {
        int k = i >> 6;
        int n = i & 63;
        sW[(k >> 1) * 128 + n * 2 + (k & 1)] = Wm[i];
    }
    __syncthreads();

    const int r0 = (int)blockIdx.x * 128 + wave * 16;
    float* sx = sX[wave];

    v8f acc0 = {}, acc1 = {}, acc2 = {}, acc3 = {};

    for (int kc = 0; kc < K; kc += 32) {
        // stage this wave's 16x32 X chunk; lane = column -> 128B coalesced rows
#pragma unroll
        for (int r = 0; r < 16; ++r) {
            int g = r0 + r;
            sx[r * 32 + lane] = (g < N) ? X[(size_t)g * K + kc + lane] : 0.0f;
        }
        // wave-local LDS ordering (DS ops are in-order; ensure data visible)
        asm volatile("s_wait_dscnt 0" ::: "memory");

#pragma unroll
        for (int k4 = 0; k4 < 32; k4 += 4) {
            const int kk = k4 + lhalf * 2;
            // A fragment: lane&15 = M, vgpr0/1 = K=kk,kk+1 (ISA 7.12.2)
            v2f a = *(const v2f*)&sx[lrow * 32 + kk];   // 8B aligned (kk even)
            // B fragments: one b64 each from interleaved sW
            const float* wp = &sW[((kc + kk) >> 1) * 128];
            v2f b0 = *(const v2f*)&wp[(lrow +  0) * 2];
            v2f b1 = *(const v2f*)&wp[(lrow + 16) * 2];
            v2f b2 = *(const v2f*)&wp[(lrow + 32) * 2];
            v2f b3 = *(const v2f*)&wp[(lrow + 48) * 2];
            acc0 = __builtin_amdgcn_wmma_f32_16x16x4_f32(false, a, false, b0,
                                                         (short)0, acc0, false, false);
            acc1 = __builtin_amdgcn_wmma_f32_16x16x4_f32(false, a, false, b1,
                                                         (short)0, acc1, false, false);
            acc2 = __builtin_amdgcn_wmma_f32_16x16x4_f32(false, a, false, b2,
                                                         (short)0, acc2, false, false);
            acc3 = __builtin_amdgcn_wmma_f32_16x16x4_f32(false, a, false, b3,
                                                         (short)0, acc3, false, false);
        }
        asm volatile("s_wait_dscnt 0" ::: "memory");  // before chunk overwrite
    }

    // D layout: vgpr r -> M = r + 8*lhalf, lane&15 -> N
#pragma unroll
    for (int r = 0; r < 8; ++r) {
        int g = r0 + r + lhalf * 8;
        if (g < N) {
            float* out = &H[(size_t)g * 64 + lrow];
            out[0]  = acc0[r];
            out[16] = acc1[r];
            out[32] = acc2[r];
            out[48] = acc3[r];
        }
    }
}

// ---------------------------------------------------------------------------
// Edge scatter: one wave per edge; Out[dst,:] += H[src,:] * dinv[src]*dinv[dst]
// Accumulator buffer is L2-resident (25.6 MB << 192 MB L2).
// Each lane: one global_load_b64 (2 contiguous channels) + 2 f32 atomics.
// ---------------------------------------------------------------------------
__global__ __launch_bounds__(256)
void gcn_scatter_kernel(const float* __restrict__ H,
                        const long long* __restrict__ srcIdx,
                        const long long* __restrict__ dstIdx,
                        const float* __restrict__ dinv,
                        float* __restrict__ Out, long long E) {
    long long e = (long long)blockIdx.x * (blockDim.x >> 5) + (threadIdx.x >> 5);
    if (e >= E) return;
    const int lane = threadIdx.x & 31;
    long long s = srcIdx[e];
    long long d = dstIdx[e];
    float nrm = dinv[s] * dinv[d];
    const float* hs = H + (size_t)s * 64;
    float* od = Out + (size_t)d * 64;
    v2f v = *(const v2f*)&hs[lane * 2];       // 8B aligned, 256B/wave coalesced
    atomicAdd(&od[lane * 2 + 0], v[0] * nrm);
    atomicAdd(&od[lane * 2 + 1], v[1] * nrm);
}

// ---------------------------------------------------------------------------
// Fused self-loop + bias + leaky-relu (float4 vectorized over channels):
//   Outp = act(Agg + Hpre * dinv^2 + b)
// (self-loop edge norm = dinv[i]*dinv[i]; folded here instead of the scatter)
// ---------------------------------------------------------------------------
__global__ __launch_bounds__(256)
void gcn_bias_act_kernel(const float* __restrict__ Agg,
                         const float* __restrict__ Hpre,
                         const float* __restrict__ dinv,
                         const float* __restrict__ bias,
                         float* __restrict__ Outp,
                         long long total4, int leaky) {
    long long t = (long long)blockIdx.x * blockDim.x + threadIdx.x;
    if (t >= total4) return;                  // total4 = N * 16 (groups of 4 ch)
    int node = (int)(t >> 4);
    int c4   = (int)(t & 15) * 4;
    float di = dinv[node];
    float s  = di * di;
    v4f a  = *(const v4f*)&Agg [t * 4];
    v4f h  = *(const v4f*)&Hpre[t * 4];
    v4f bb = *(const v4f*)&bias[c4];
    v4f v;
#pragma unroll
    for (int j = 0; j < 4; ++j) {
        float x = a[j] + h[j] * s + bb[j];
        if (leaky) x = (x >= 0.0f) ? x : NEG_SLOPE * x;
        v[j] = x;
    }
    *(v4f*)&Outp[t * 4] = v;
}

// ---------------------------------------------------------------------------
extern "C" void kernel_launch(void* const* d_in, const int* in_sizes, int n_in,
                              void* d_out, int out_size, void* d_ws, size_t ws_size,
                              hipStream_t stream) {
    const float*     x    = (const float*)d_in[0];
    const long long* ei   = (const long long*)d_in[1];   // int64, shape (2,E)
    const float*     W1   = (const float*)d_in[2];
    const float*     b1   = (const float*)d_in[3];
    const float*     W2   = (const float*)d_in[4];
    const float*     b2   = (const float*)d_in[5];
    const float*     W3   = (const float*)d_in[6];
    const float*     b3   = (const float*)d_in[7];
    float* out = (float*)d_out;

    const int       N  = in_sizes[0] / 128;
    const long long E  = (long long)in_sizes[1] / 2;
    const long long NC = (long long)N * 64;
    const long long N4 = (long long)N * 16;   // float4 groups per feature map

    const long long* srcI = ei;        // row 0
    const long long* dstI = ei + E;    // row 1

    // workspace layout (floats): dinv[N] | bufA[N*64] | bufB[N*64]
    float* dinv = (float*)d_ws;
    float* bufA = dinv + ((N + 63) & ~63);
    float* bufB = bufA + NC;

    const int T = 256;
    const int gN    = (N + T - 1) / T;
    const int gE    = (int)((E + T - 1) / T);
    const int gEw   = (int)((E + 7) / 8);           // one wave (32 lanes) per edge
    const int gN4   = (int)((N4 + T - 1) / T);
    const int gGemm = (N + 127) / 128;

    // --- normalization ---
    deg_init_kernel <<<gN, T, 0, stream>>>(dinv, N);
    deg_count_kernel<<<gE, T, 0, stream>>>(dstI, dinv, E);
    deg_rsqrt_kernel<<<gN, T, 0, stream>>>(dinv, N);

    // --- layer 1: h1 = x @ W1 ; agg ; a1 = leaky(agg + self + b1) ---
    gcn_gemm_wmma_kernel<<<gGemm, T, 0, stream>>>(x, W1, bufA, N, 128);
    hipMemsetAsync(bufB, 0, NC * sizeof(float), stream);
    gcn_scatter_kernel <<<gEw, T, 0, stream>>>(bufA, srcI, dstI, dinv, bufB, E);
    gcn_bias_act_kernel<<<gN4, T, 0, stream>>>(bufB, bufA, dinv, b1, bufA, N4, 1);

    // --- layer 2 ---
    gcn_gemm_wmma_kernel<<<gGemm, T, 0, stream>>>(bufA, W2, bufB, N, 64);
    hipMemsetAsync(bufA, 0, NC * sizeof(float), stream);
    gcn_scatter_kernel <<<gEw, T, 0, stream>>>(bufB, srcI, dstI, dinv, bufA, E);
    gcn_bias_act_kernel<<<gN4, T, 0, stream>>>(bufA, bufB, dinv, b2, bufB, N4, 1);

    // --- layer 3 (no activation) ---
    gcn_gemm_wmma_kernel<<<gGemm, T, 0, stream>>>(bufB, W3, bufA, N, 64);
    hipMemsetAsync(out, 0, NC * sizeof(float), stream);
    gcn_scatter_kernel <<<gEw, T, 0, stream>>>(bufA, srcI, dstI, dinv, out, E);
    gcn_bias_act_kernel<<<gN4, T, 0, stream>>>(out, bufA, dinv, b3, out, N4, 0);
}